// GraphEmbedding_9929964388984
// MI455X (gfx1250) — compile-verified
//
#include <hip/hip_runtime.h>
#include <hip/hip_bf16.h>
#include <math.h>

// Problem dims (fixed by the reference)
#define NNODES 200000
#define NEDGES 1000000
#define BB     4096
#define KK     20
#define DN     128
#define DEMB   128
#define DMEM   256   // DN + DEMB
#define DT     128
#define DE     128
#define NBINS  300

typedef __attribute__((ext_vector_type(2))) float v2f;
typedef __attribute__((ext_vector_type(4))) float v4f;
typedef __attribute__((ext_vector_type(8))) float v8f;

__device__ __forceinline__ v4f fma4(float s, v4f a, v4f b) {
    v4f r;
    r.x = fmaf(s, a.x, b.x); r.y = fmaf(s, a.y, b.y);
    r.z = fmaf(s, a.z, b.z); r.w = fmaf(s, a.w, b.w);
    return r;
}
__device__ __forceinline__ v4f add4(v4f a, v4f b) {
    v4f r; r.x = a.x + b.x; r.y = a.y + b.y; r.z = a.z + b.z; r.w = a.w + b.w; return r;
}
__device__ __forceinline__ v4f mul4s(v4f a, float s) {
    v4f r; r.x = a.x * s; r.y = a.y * s; r.z = a.z * s; r.w = a.w * s; return r;
}

// searchsorted(bounds, x, side='left') - 1, clamped to [0, NBINS-1]
__device__ __forceinline__ int bucketize(const float* __restrict__ bounds, float x) {
    int lo = 0, hi = NBINS + 1;           // 301 boundary entries
    #pragma unroll 1
    while (lo < hi) {
        int mid = (lo + hi) >> 1;
        if (bounds[mid] < x) lo = mid + 1; else hi = mid;
    }
    int idx = lo - 1;
    idx = idx < 0 ? 0 : idx;
    idx = idx > (NBINS - 1) ? (NBINS - 1) : idx;
    return idx;
}

// Kernel 1: gather + masked K-reduction, float4 (b128) data path.
// Block = 256 threads = 4 batch rows x 64 lanes; each lane owns 4 consecutive dims.
// Writes out[b,:] = src0 + mean_k(nb0*valid), xbar[b,:] = mean_k(concat(ete,ef)*valid).
__global__ __launch_bounds__(256) void gather_reduce_kernel(
    const float* __restrict__ nodef,   // [N, DN]
    const float* __restrict__ edgef,   // [E, DE]
    const float* __restrict__ mem,     // [N, DMEM]
    const float* __restrict__ itab,    // [NBINS, DEMB]
    const float* __restrict__ bounds,  // [NBINS+1]
    const float* __restrict__ tw,      // [DT]
    const float* __restrict__ tb,      // [DT]
    const float* __restrict__ ts,      // [B]
    const float* __restrict__ etimes,  // [B, K]
    const float* __restrict__ ivals,   // [B]
    const float* __restrict__ nivals,  // [B, K]
    const int*   __restrict__ srcn,    // [B]
    const int*   __restrict__ nbrs,    // [B, K]
    const int*   __restrict__ eidx,    // [B, K]
    float* __restrict__ out,           // [B, DMEM]
    float* __restrict__ xbar)          // [B, DMEM]
{
    __shared__ float sb[NBINS + 1];
    const int tid = threadIdx.x;
    for (int i = tid; i < NBINS + 1; i += 256) sb[i] = bounds[i];
    __syncthreads();

    const int rl = tid >> 6;               // row within block (0..3)
    const int t  = tid & 63;               // lane within row
    const int b  = (blockIdx.x << 2) + rl;
    const int d0 = t << 2;                 // feature start (0,4,...,252)
    const bool lo = (d0 < DT);             // wave-uniform split
    const int dh0 = lo ? d0 : (d0 - DT);

    v4f twv = {0.f,0.f,0.f,0.f}, tbv = {0.f,0.f,0.f,0.f};
    if (lo) { twv = *(const v4f*)(tw + dh0); tbv = *(const v4f*)(tb + dh0); }

    // source embedding
    const int s = srcn[b];
    v4f srcv = *(const v4f*)(mem + (long)s * DMEM + d0);
    if (lo) srcv = add4(srcv, *(const v4f*)(nodef + (long)s * DN + d0));
    else    srcv = add4(srcv, *(const v4f*)(itab + bucketize(sb, ivals[b]) * DEMB + dh0));

    const float t0 = ts[b];
    v4f accn = {0.f,0.f,0.f,0.f}, accx = {0.f,0.f,0.f,0.f};
    float cnt = 0.f;

    #pragma unroll 4
    for (int k = 0; k < KK; ++k) {
        const int nb = nbrs[b * KK + k];
        const float valid = (nb != -1) ? 1.f : 0.f;
        cnt += valid;
        const int nbc = nb < 0 ? 0 : nb;

        v4f nv = *(const v4f*)(mem + (long)nbc * DMEM + d0);   // b128 row gather
        v4f xv;
        if (lo) {
            nv = add4(nv, *(const v4f*)(nodef + (long)nbc * DN + d0));
            const float delta = t0 - etimes[b * KK + k];
            xv.x = __cosf(fmaf(delta, twv.x, tbv.x));
            xv.y = __cosf(fmaf(delta, twv.y, tbv.y));
            xv.z = __cosf(fmaf(delta, twv.z, tbv.z));
            xv.w = __cosf(fmaf(delta, twv.w, tbv.w));
        } else {
            nv = add4(nv, *(const v4f*)(itab + bucketize(sb, nivals[b * KK + k]) * DEMB + dh0));
            int e = eidx[b * KK + k]; e = e < 0 ? 0 : e;
            xv = *(const v4f*)(edgef + (long)e * DE + dh0);
        }
        accn = fma4(valid, nv, accn);
        accx = fma4(valid, xv, accx);
    }

    const float inv = 1.f / fmaxf(cnt, 1.f);
    *(v4f*)(out  + (long)b * DMEM + d0) = add4(srcv, mul4s(accn, inv));
    *(v4f*)(xbar + (long)b * DMEM + d0) = mul4s(accx, inv);
}

// Kernel 2: out += xbar @ W_agg  (M=4096, K=256, N=256, fp32 WMMA 16x16x4).
// Each wave computes FOUR 16x16 M-tiles for one N-tile: the B fragment is
// loaded once per K-step and reused by 4 WMMAs (load:wmma ratio 1.5:1).
#define MT 4
__global__ __launch_bounds__(128) void gemm_msg_kernel(
    const float* __restrict__ xbar,   // [B, DMEM]
    const float* __restrict__ wagg,   // [DMEM, DMEM] row-major (f, d)
    float* __restrict__ out)          // [B, DMEM]
{
    const int lane  = threadIdx.x & 31;
    const int wave  = threadIdx.x >> 5;             // 4 N-tiles per block
    const int tileM = blockIdx.x * (16 * MT);
    const int tileN = (blockIdx.y * 4 + wave) * 16;
    const int half  = lane >> 4;                    // 0: K pair {0,1}; 1: {2,3}
    const int l     = lane & 15;

    const float* __restrict__ abase = xbar + (tileM + l) * DMEM + 2 * half;
    const float* __restrict__ bcol  = wagg + tileN + l;

    v8f c0 = {0.f,0.f,0.f,0.f,0.f,0.f,0.f,0.f};
    v8f c1 = c0, c2 = c0, c3 = c0;

    #pragma unroll 4
    for (int kk = 0; kk < DMEM; kk += 4) {
        v2f bfrag;
        bfrag.x = bcol[(kk + 2 * half + 0) * DMEM];
        bfrag.y = bcol[(kk + 2 * half + 1) * DMEM];
        v2f a0 = *(const v2f*)(abase + kk);
        v2f a1 = *(const v2f*)(abase + 16 * DMEM + kk);
        v2f a2 = *(const v2f*)(abase + 32 * DMEM + kk);
        v2f a3 = *(const v2f*)(abase + 48 * DMEM + kk);
        c0 = __builtin_amdgcn_wmma_f32_16x16x4_f32(false, a0, false, bfrag, (short)0, c0, false, false);
        c1 = __builtin_amdgcn_wmma_f32_16x16x4_f32(false, a1, false, bfrag, (short)0, c1, false, false);
        c2 = __builtin_amdgcn_wmma_f32_16x16x4_f32(false, a2, false, bfrag, (short)0, c2, false, false);
        c3 = __builtin_amdgcn_wmma_f32_16x16x4_f32(false, a3, false, bfrag, (short)0, c3, false, false);
    }

    // C/D layout: VGPR i -> (M = tile + i + 8*half, N = tileN + l)
    #pragma unroll
    for (int i = 0; i < 8; ++i) {
        const int col = tileN + l;
        out[(tileM +  0 + i + 8 * half) * DMEM + col] += c0[i];
        out[(tileM + 16 + i + 8 * half) * DMEM + col] += c1[i];
        out[(tileM + 32 + i + 8 * half) * DMEM + col] += c2[i];
        out[(tileM + 48 + i + 8 * half) * DMEM + col] += c3[i];
    }
}

extern "C" void kernel_launch(void* const* d_in, const int* in_sizes, int n_in,
                              void* d_out, int out_size, void* d_ws, size_t ws_size,
                              hipStream_t stream) {
    const float* nodef  = (const float*)d_in[0];
    const float* edgef  = (const float*)d_in[1];
    const float* mem    = (const float*)d_in[2];
    const float* itab   = (const float*)d_in[3];
    const float* bounds = (const float*)d_in[4];
    const float* tw     = (const float*)d_in[5];
    const float* tb     = (const float*)d_in[6];
    const float* wagg   = (const float*)d_in[7];
    const float* ts     = (const float*)d_in[8];
    const float* etimes = (const float*)d_in[9];
    const float* ivals  = (const float*)d_in[10];
    const float* nivals = (const float*)d_in[11];
    const int*   srcn   = (const int*)d_in[12];
    const int*   nbrs   = (const int*)d_in[13];
    const int*   eidx   = (const int*)d_in[14];

    float* out  = (float*)d_out;
    float* xbar = (float*)d_ws;    // B*DMEM floats = 4 MB scratch

    gather_reduce_kernel<<<BB / 4, 256, 0, stream>>>(
        nodef, edgef, mem, itab, bounds, tw, tb, ts, etimes,
        ivals, nivals, srcn, nbrs, eidx, out, xbar);

    dim3 g2(BB / (16 * MT), DMEM / (16 * 4));   // (64, 4) blocks, 4 waves each
    gemm_msg_kernel<<<g2, 128, 0, stream>>>(xbar, wagg, out);
}